// BipartiteGCN_54030688583922
// MI455X (gfx1250) — compile-verified
//
#include <hip/hip_runtime.h>

#define D 128

typedef __attribute__((ext_vector_type(2))) float v2f;
typedef __attribute__((ext_vector_type(8))) float v8f;

// Hardware FP32 atomic add: relaxed, device scope, result discarded ->
// non-returning global_atomic_add_f32 (no CAS loop).
__device__ __forceinline__ void atom_add_f32(float* p, float v) {
  __hip_atomic_fetch_add(p, v, __ATOMIC_RELAXED, __HIP_MEMORY_SCOPE_AGENT);
}

// ---------------- zero fill ----------------
__global__ __launch_bounds__(256) void zero_f4(float4* __restrict__ p, int n4) {
  int i = blockIdx.x * blockDim.x + threadIdx.x;
  int stride = gridDim.x * blockDim.x;
  float4 z; z.x = z.y = z.z = z.w = 0.0f;
  for (; i < n4; i += stride) p[i] = z;
}

// ---------------- bidirectional edge scatter-add (wave32 per edge) ----------------
__global__ __launch_bounds__(256) void scatter_bi(
    const int* __restrict__ ei, int E,
    const float* __restrict__ xa, const float* __restrict__ xp,
    float* Sp, float* Sa, float* degP, float* degA)
{
  int wid = (blockIdx.x * blockDim.x + threadIdx.x) >> 5;
  int lane = threadIdx.x & 31;
  if (wid >= E) return;
  int a = ei[wid];
  int p = ei[E + wid];
  const float* ra = xa + (size_t)a * D;
  const float* rp = xp + (size_t)p * D;
  float* wp = Sp + (size_t)p * D;
  float* wa = Sa + (size_t)a * D;
#pragma unroll
  for (int i = 0; i < 4; ++i) {
    int c = lane + 32 * i;
    atom_add_f32(&wp[c], ra[c]);
    atom_add_f32(&wa[c], rp[c]);
  }
  if (lane == 0) { atom_add_f32(&degP[p], 1.0f); atom_add_f32(&degA[a], 1.0f); }
}

// ---------------- unidirectional edge scatter-add ----------------
__global__ __launch_bounds__(256) void scatter_uni(
    const int* __restrict__ ei, int E,
    const float* __restrict__ x, float* S, float* deg)
{
  int wid = (blockIdx.x * blockDim.x + threadIdx.x) >> 5;
  int lane = threadIdx.x & 31;
  if (wid >= E) return;
  int s = ei[wid];
  int d = ei[E + wid];
  const float* rs = x + (size_t)s * D;
  float* wd = S + (size_t)d * D;
#pragma unroll
  for (int i = 0; i < 4; ++i) {
    int c = lane + 32 * i;
    atom_add_f32(&wd[c], rs[c]);
  }
  if (lane == 0) atom_add_f32(&deg[d], 1.0f);
}

// ---------------- fused dual GEMM: out = A1@W1^T + A2@W2^T + deg*b1 + b2 (+ Cadd) ----
// torch Linear convention: y[i,j] = sum_k x[i,k]*W[j,k] + b[j]
// WMMA f32 16x16x4: A frag = float2 {A[r][kb+kh], A[r][kb+kh+1]}, r=lane&15, kh=2*(lane>>4)
//                   B frag = float2 {B[kb+kh][col], ...} = {W[col][kb+kh], W[col][kb+kh+1]}
// C tile: c[i] -> row = i + 8*(lane>>4), col = lane&15.
// out may alias A1 (in-place): all A1 reads retire before __syncthreads, stores after.
__global__ __launch_bounds__(256) void gemm_fused(
    const float* A1, const float* __restrict__ W1, const float* __restrict__ b1,
    const float* __restrict__ deg,
    const float* A2, const float* __restrict__ W2, const float* __restrict__ b2,
    const float* Cadd,
    float* out, int M)
{
  const int lane  = threadIdx.x & 31;
  const int wave  = threadIdx.x >> 5;       // 0..7 -> column tile
  const int mbase = blockIdx.x * 16;
  const int nbase = wave * 16;
  const int r     = lane & 15;
  const int kh    = (lane >> 4) * 2;        // 0 or 2

  v8f c = {};
  {
    const float* ap = A1 + (size_t)(mbase + r) * D + kh;
    const float* wp = W1 + (size_t)(nbase + r) * D + kh;
#pragma unroll
    for (int kb = 0; kb < D; kb += 4) {
      v2f a = *(const v2f*)(ap + kb);
      v2f b = *(const v2f*)(wp + kb);
      c = __builtin_amdgcn_wmma_f32_16x16x4_f32(false, a, false, b,
                                                (short)0, c, false, false);
    }
  }
  if (A2) {
    const float* ap = A2 + (size_t)(mbase + r) * D + kh;
    const float* wp = W2 + (size_t)(nbase + r) * D + kh;
#pragma unroll
    for (int kb = 0; kb < D; kb += 4) {
      v2f a = *(const v2f*)(ap + kb);
      v2f b = *(const v2f*)(wp + kb);
      c = __builtin_amdgcn_wmma_f32_16x16x4_f32(false, a, false, b,
                                                (short)0, c, false, false);
    }
  }

  __syncthreads();  // all A1 reads (consumed by WMMA) done before in-place stores

  const int col    = nbase + (lane & 15);
  const int rowoff = (lane >> 4) * 8;
  const float bb1 = b1[col];
  const float bb2 = b2 ? b2[col] : 0.0f;
#pragma unroll
  for (int i = 0; i < 8; ++i) {
    int row = mbase + rowoff + i;
    float v = c[i] + deg[row] * bb1 + bb2;
    if (Cadd) v += Cadd[(size_t)row * D + col];
    out[(size_t)row * D + col] = v;
  }
}

// ---------------- supervision dot products (wave32 per edge) ----------------
__global__ __launch_bounds__(256) void dot_rows(
    const int* __restrict__ se, int E,
    const float* __restrict__ xa, const float* __restrict__ xp,
    float* __restrict__ out)
{
  int wid = (blockIdx.x * blockDim.x + threadIdx.x) >> 5;
  int lane = threadIdx.x & 31;
  if (wid >= E) return;
  int a = se[wid];
  int p = se[E + wid];
  const float* ra = xa + (size_t)a * D;
  const float* rp = xp + (size_t)p * D;
  float s = 0.0f;
#pragma unroll
  for (int i = 0; i < 4; ++i) {
    int c = lane + 32 * i;
    s += ra[c] * rp[c];
  }
#pragma unroll
  for (int off = 16; off > 0; off >>= 1) s += __shfl_down(s, off, 32);
  if (lane == 0) out[wid] = s;
}

extern "C" void kernel_launch(void* const* d_in, const int* in_sizes, int n_in,
                              void* d_out, int out_size, void* d_ws, size_t ws_size,
                              hipStream_t stream) {
  const float* x_author = (const float*)d_in[0];
  const float* x_paper  = (const float*)d_in[1];
  const int*   edge     = (const int*)d_in[2];
  const int*   co       = (const int*)d_in[3];
  const int*   sup      = (const int*)d_in[4];
  const float* W_a2p   = (const float*)d_in[5];
  const float* b_a2p   = (const float*)d_in[6];
  const float* W_p2a   = (const float*)d_in[7];
  const float* b_p2a   = (const float*)d_in[8];
  const float* W_aself = (const float*)d_in[9];
  const float* b_aself = (const float*)d_in[10];
  const float* W_pself = (const float*)d_in[11];
  const float* b_pself = (const float*)d_in[12];
  const float* W_co    = (const float*)d_in[13];
  const float* b_co    = (const float*)d_in[14];

  const int NA   = in_sizes[0] / D;   // 100000
  const int NP   = in_sizes[1] / D;   // 200000
  const int E    = in_sizes[2] / 2;   // 500000
  const int Eco  = in_sizes[3] / 2;   // 250000
  const int Esup = in_sizes[4] / 2;   // 100000
  const int DD   = D * D;

  float* bufPA = (float*)d_ws;
  float* bufPB = bufPA + (size_t)NP * D;
  float* bufA0 = bufPB + (size_t)NP * D;
  float* bufA1 = bufA0 + (size_t)NA * D;
  float* degP  = bufA1 + (size_t)NA * D;
  float* degA  = degP + NP;
  size_t need = (size_t)((char*)(degA + NA) - (char*)d_ws);
  if (need > ws_size) return;

  auto zero = [&](float* p, size_t nf) {
    int n4 = (int)(nf / 4);
    int blocks = (n4 + 255) / 256;
    if (blocks > 8192) blocks = 8192;
    zero_f4<<<blocks, 256, 0, stream>>>((float4*)p, n4);
  };

  const float* xa = x_author;
  const float* xp = x_paper;
  float* paperBuf[2] = { bufPA, bufPB };

  for (int l = 0; l < 2; ++l) {
    float* Sp = paperBuf[l];
    float* Sa = bufA0;
    zero(Sp, (size_t)NP * D);
    zero(Sa, (size_t)NA * D);
    zero(degP, NP);
    zero(degA, NA);
    scatter_bi<<<(E + 7) / 8, 256, 0, stream>>>(edge, E, xa, xp, Sp, Sa, degP, degA);

    // x_paper' = Sp@W_a2p^T + degP*b_a2p + x_paper@W_pself^T + b_pself   (in place into Sp)
    gemm_fused<<<NP / 16, 256, 0, stream>>>(Sp, W_a2p + l * DD, b_a2p + l * D, degP,
                                            xp, W_pself + l * DD, b_pself + l * D,
                                            nullptr, Sp, NP);
    // x_author_mid = Sa@W_p2a^T + degA*b_p2a + x_author@W_aself^T + b_aself  (into Sa)
    gemm_fused<<<NA / 16, 256, 0, stream>>>(Sa, W_p2a + l * DD, b_p2a + l * D, degA,
                                            xa, W_aself + l * DD, b_aself + l * D,
                                            nullptr, Sa, NA);

    // coauthor pass: Sco (bufA1) accumulates x_mid rows; reuse degA as degCo
    zero(bufA1, (size_t)NA * D);
    zero(degA, NA);
    scatter_uni<<<(Eco + 7) / 8, 256, 0, stream>>>(co, Eco, Sa, bufA1, degA);
    // x_author' = x_mid + Sco@W_co^T + degCo*b_co   (in place into bufA1, Cadd = x_mid)
    gemm_fused<<<NA / 16, 256, 0, stream>>>(bufA1, W_co + l * DD, b_co + l * D, degA,
                                            nullptr, nullptr, nullptr,
                                            Sa, bufA1, NA);
    xa = bufA1;
    xp = Sp;
  }

  dot_rows<<<(Esup + 7) / 8, 256, 0, stream>>>(sup, Esup, xa, xp, (float*)d_out);
}